// DART_19232863551644
// MI455X (gfx1250) — compile-verified
//
#include <hip/hip_runtime.h>
#include <math.h>
#include <stdint.h>

typedef float v2f __attribute__((ext_vector_type(2)));
typedef float v8f __attribute__((ext_vector_type(8)));

// ---------------------------------------------------------------------------
// Helpers
// ---------------------------------------------------------------------------

__device__ __forceinline__ int mod255(int v) {
    if (v >= 255) v -= 255;
    if (v >= 255) v -= 255;
    return v;
}

// CDNA5 async copy: global -> LDS, 16B per lane, tracked by ASYNCcnt.
// Generic pointers to LDS carry the LDS byte offset in their low 32 bits.
__device__ __forceinline__ void async_copy_b128(const float* g, float* l) {
    const uint32_t loff = (uint32_t)(uintptr_t)l;
    asm volatile("global_load_async_to_lds_b128 %0, %1, off"
                 :: "v"(loff), "v"(g)
                 : "memory");
}

#define ASYNC_WAIT(n) asm volatile("s_wait_asynccnt " #n ::: "memory")

// ---------------------------------------------------------------------------
// Masked GEMM via V_WMMA_F32_16X16X4_F32.
//   C[m, n] = act( sum_k A[m,k] * mask(n,k) * W[n,k] + bias[n] )
// A: M x Kdim row-major, W: N x Kdim row-major, C: M x N row-major.
// Block: 256 threads = 8 waves -> 64x64 C macro-tile, Kc = 32.
// Wave tile: 16(M) x 32(N) -> two accumulators sharing one A fragment.
// A/W chunks staged in LDS via global_load_async_to_lds_b128, double-buffered.
// MODE 0 (m0):  mask = (n % 255) >= k
// MODE 1 (mh):  mask = (n % 255) >= (k % 255)
// MODE 2 (ml):  mask = ((n >> 5) - 1) >= (k % 255)   (signed; d=0 row all-zero)
// ---------------------------------------------------------------------------

template <int MODE, bool RELU>
__global__ __launch_bounds__(256) void masked_gemm_wmma(
    const float* __restrict__ A,
    const float* __restrict__ W,
    const float* __restrict__ bias,
    float* __restrict__ C,
    int M, int N, int Kdim)
{
    constexpr int LDA = 36;   // padded row stride (floats): 144B, 16B aligned,
                              // conflict-free ds_load_b64 (36r mod 64 distinct)
    __shared__ __align__(16) float Ash[2][64 * LDA];
    __shared__ __align__(16) float Wsh[2][64 * LDA];

    const int gn = N >> 6;                    // N macro-tiles of 64
    const int mt = (blockIdx.x / gn) << 6;    // macro-tile M origin
    const int nt = (blockIdx.x % gn) << 6;    // macro-tile N origin

    const int t    = (int)threadIdx.x;
    const int lane = t & 31;
    const int w    = t >> 5;                  // wave 0..7
    const int wm   = w >> 1;                  // 0..3 (M sub-tile of 16)
    const int wn   = w & 1;                   // 0..1 (N sub-tile of 32)
    const int half = lane >> 4;               // K pair select for WMMA A/B
    const int lr   = lane & 15;

    // Copy-lane assignment: 256 threads cover one Kc=32 chunk with b128 ops.
    const int crow = t >> 3;                  // 0..31
    const int ccol = (t & 7) << 2;            // 0,4,...,28
    const float* gA0 = A + (size_t)(mt + crow) * Kdim + ccol;
    const float* gA1 = A + (size_t)(mt + crow + 32) * Kdim + ccol;
    const float* gW0 = W + (size_t)(nt + crow) * Kdim + ccol;
    const float* gW1 = W + (size_t)(nt + crow + 32) * Kdim + ccol;
    const int lof = crow * LDA + ccol;

    auto issue_copy = [&](int k0, int buf) {
        async_copy_b128(gA0 + k0, &Ash[buf][lof]);
        async_copy_b128(gA1 + k0, &Ash[buf][lof + 32 * LDA]);
        async_copy_b128(gW0 + k0, &Wsh[buf][lof]);
        async_copy_b128(gW1 + k0, &Wsh[buf][lof + 32 * LDA]);
    };

    const int ncol0 = nt + (wn << 5) + lr;    // first N fragment column
    const int ncol1 = ncol0 + 16;             // second N fragment column
    const int ndeg0 = (MODE == 2) ? ((ncol0 >> 5) - 1) : mod255(ncol0);
    const int ndeg1 = (MODE == 2) ? ((ncol1 >> 5) - 1) : mod255(ncol1);

    v8f acc0 = {};
    v8f acc1 = {};
    issue_copy(0, 0);

    const int nChunks = Kdim >> 5;
    for (int c = 0; c < nChunks; ++c) {
        const int buf = c & 1;
        if (c + 1 < nChunks) {
            issue_copy((c + 1) << 5, buf ^ 1);
            ASYNC_WAIT(0x4);                  // chunk c landed; c+1 in flight
        } else {
            ASYNC_WAIT(0x0);
        }
        __syncthreads();

        const int k0 = c << 5;
        const float* As  = &Ash[buf][((wm << 4) + lr) * LDA];
        const float* Ws0 = &Wsh[buf][((wn << 5) + lr) * LDA];
        const float* Ws1 = Ws0 + 16 * LDA;

        v2f avs[8], wv0s[8], wv1s[8];
#pragma unroll
        for (int u = 0; u < 8; ++u) {
            const int ka = (u << 2) + (half << 1);
            avs[u]  = *(const v2f*)(As  + ka);
            wv0s[u] = *(const v2f*)(Ws0 + ka);
            wv1s[u] = *(const v2f*)(Ws1 + ka);
        }
#pragma unroll
        for (int u = 0; u < 8; ++u) {
            const int k = k0 + (u << 2) + (half << 1);
            int ka0, kb0;
            if (MODE == 0) { ka0 = k;         kb0 = k + 1; }
            else           { ka0 = mod255(k); kb0 = mod255(k + 1); }

            v2f bv0, bv1;
            bv0.x = wv0s[u].x * ((ndeg0 >= ka0) ? 1.0f : 0.0f);
            bv0.y = wv0s[u].y * ((ndeg0 >= kb0) ? 1.0f : 0.0f);
            bv1.x = wv1s[u].x * ((ndeg1 >= ka0) ? 1.0f : 0.0f);
            bv1.y = wv1s[u].y * ((ndeg1 >= kb0) ? 1.0f : 0.0f);

            acc0 = __builtin_amdgcn_wmma_f32_16x16x4_f32(
                false, avs[u], false, bv0, (short)0, acc0, false, false);
            acc1 = __builtin_amdgcn_wmma_f32_16x16x4_f32(
                false, avs[u], false, bv1, (short)0, acc1, false, false);
        }
        __syncthreads();                      // buf free for chunk c+2 issue
    }

    const float bn0 = bias[ncol0];
    const float bn1 = bias[ncol1];
    const int rowBase = mt + (wm << 4) + (half << 3);
#pragma unroll
    for (int r = 0; r < 8; ++r) {
        float v0 = acc0[r] + bn0;
        float v1 = acc1[r] + bn1;
        if (RELU) { v0 = fmaxf(v0, 0.0f); v1 = fmaxf(v1, 0.0f); }
        float* crow_ptr = C + (size_t)(rowBase + r) * N;
        crow_ptr[ncol0] = v0;
        crow_ptr[ncol1] = v1;
    }
}

// ---------------------------------------------------------------------------
// Per-sample autoregressive logsumexp scan over the 4-state chain.
// theta[b,d,c,i,j] is 32 contiguous floats per (b,d).
// ---------------------------------------------------------------------------

__device__ __forceinline__ void lpa4(const float* __restrict__ ulpa, int d, float* l)
{
    const float u0 = ulpa[d * 4 + 0], u1 = ulpa[d * 4 + 1];
    const float u2 = ulpa[d * 4 + 2], u3 = ulpa[d * 4 + 3];
    const float mx = fmaxf(fmaxf(u0, u1), fmaxf(u2, u3));
    const float s  = expf(u0 - mx) + expf(u1 - mx) + expf(u2 - mx) + expf(u3 - mx);
    const float lse = mx + logf(s);
    l[0] = u0 - lse; l[1] = u1 - lse; l[2] = u2 - lse; l[3] = u3 - lse;
}

__global__ __launch_bounds__(256) void dart_scan(
    const float* __restrict__ x,      // B x 256
    const float* __restrict__ theta,  // B x 8192
    const float* __restrict__ ulpa,   // 255 x 4
    float* __restrict__ logp,         // B
    int B)
{
    const int b = blockIdx.x * blockDim.x + (int)threadIdx.x;
    if (b >= B) return;

    const float HALF_LOG_2PI = 0.91893853320467274178f;
    const float* th = theta + (size_t)b * 8192;
    const float* xr = x + (size_t)b * 256;

    float alpha[4];
    {   // d = 0: first = joint[:, 0, row i=0]
        float l[4];
        lpa4(ulpa, 0, l);
        const float xe = xr[0];
        const float* t = th;
#pragma unroll
        for (int j = 0; j < 4; ++j) {
            const float mu = t[j];                    // c=0, i=0, j
            const float sd = expf(t[16 + j]) + 0.01f; // c=1, i=0, j
            const float z  = (xe - mu) / sd;
            alpha[j] = -0.5f * z * z - logf(sd) - HALF_LOG_2PI + l[j];
        }
    }

    for (int d = 1; d <= 254; ++d) {
        float l[4];
        lpa4(ulpa, d, l);
        const float xe = xr[d];
        const float* t = th + d * 32;
        float na[4];
#pragma unroll
        for (int j = 0; j < 4; ++j) {
            float v[4];
            float mx = -3.0e38f;
#pragma unroll
            for (int i = 0; i < 4; ++i) {
                const float mu = t[i * 4 + j];
                const float sd = expf(t[16 + i * 4 + j]) + 0.01f;
                const float z  = (xe - mu) / sd;
                const float s  = alpha[i] - 0.5f * z * z - logf(sd) - HALF_LOG_2PI;
                v[i] = s;
                mx = fmaxf(mx, s);
            }
            const float ssum = expf(v[0] - mx) + expf(v[1] - mx)
                             + expf(v[2] - mx) + expf(v[3] - mx);
            na[j] = mx + logf(ssum) + l[j];
        }
        alpha[0] = na[0]; alpha[1] = na[1]; alpha[2] = na[2]; alpha[3] = na[3];
    }

    {   // last: d = 255, column j = 0
        const float xe = xr[255];
        const float* t = th + 255 * 32;
        float v[4];
        float mx = -3.0e38f;
#pragma unroll
        for (int i = 0; i < 4; ++i) {
            const float mu = t[i * 4];
            const float sd = expf(t[16 + i * 4]) + 0.01f;
            const float z  = (xe - mu) / sd;
            const float s  = alpha[i] - 0.5f * z * z - logf(sd) - HALF_LOG_2PI;
            v[i] = s;
            mx = fmaxf(mx, s);
        }
        const float ssum = expf(v[0] - mx) + expf(v[1] - mx)
                         + expf(v[2] - mx) + expf(v[3] - mx);
        logp[b] = mx + logf(ssum);
    }
}

// ---------------------------------------------------------------------------

extern "C" void kernel_launch(void* const* d_in, const int* in_sizes, int n_in,
                              void* d_out, int out_size, void* d_ws, size_t ws_size,
                              hipStream_t stream)
{
    const float* x    = (const float*)d_in[0];
    const float* W0   = (const float*)d_in[1];
    const float* b0   = (const float*)d_in[2];
    const float* W1   = (const float*)d_in[3];
    const float* b1   = (const float*)d_in[4];
    const float* W2   = (const float*)d_in[5];
    const float* b2   = (const float*)d_in[6];
    const float* Wout = (const float*)d_in[7];
    const float* bout = (const float*)d_in[8];
    const float* ulpa = (const float*)d_in[9];
    float* out = (float*)d_out;

    const int B = 1024, D = 256, H = 512, K = 32;

    float* h0 = (float*)d_ws;                 // B x H
    float* h1 = h0 + (size_t)B * H;           // B x H
    float* theta = out + B;                   // B x (D*K), after log_p block

    const int blkH  = (B / 64) * (H / 64);        // 128 blocks
    const int blkTh = (B / 64) * ((D * K) / 64);  // 2048 blocks

    masked_gemm_wmma<0, true><<<blkH, 256, 0, stream>>>(x,  W0,   b0,   h0,    B, H,     D);
    masked_gemm_wmma<1, true><<<blkH, 256, 0, stream>>>(h0, W1,   b1,   h1,    B, H,     H);
    masked_gemm_wmma<1, true><<<blkH, 256, 0, stream>>>(h1, W2,   b2,   h0,    B, H,     H);
    masked_gemm_wmma<2, false><<<blkTh, 256, 0, stream>>>(h0, Wout, bout, theta, B, D * K, H);

    dart_scan<<<(B + 255) / 256, 256, 0, stream>>>(x, theta, ulpa, out, B);
}